// GATv2_39015482917698
// MI455X (gfx1250) — compile-verified
//
#include <hip/hip_runtime.h>
#include <hip/hip_bf16.h>
#include <math.h>

// ---------------------------------------------------------------------------
// GATv2 (3 layers, H=8, C=8, HC=64) + graph softmax readout for MI455X gfx1250.
// Dense node GEMMs use V_WMMA_F32_16X16X4_F32; edge phase is fused gather /
// segment-softmax / scatter-add kernels (atomic f32).
// ---------------------------------------------------------------------------

typedef __attribute__((ext_vector_type(2))) float v2f;
typedef __attribute__((ext_vector_type(8))) float v8f;

#define HC 64
#define HEADS 8
#define CPH 8
#define FE 4
#define NGRAPH 64
#define NCLS 4
#define NEG_SLOPE 0.2f
#define BN_EPS 1e-5f

__device__ __forceinline__ float leaky(float v) {
  return v > 0.0f ? v : NEG_SLOPE * v;
}

__device__ __forceinline__ void atomicMaxF(float* addr, float v) {
  unsigned int* ua = (unsigned int*)addr;
  unsigned int old = *ua;
  while (true) {
    float f = __uint_as_float(old);
    if (f >= v) break;
    unsigned int assumed = old;
    old = atomicCAS(ua, assumed, __float_as_uint(v));
    if (old == assumed) break;
  }
}

// ------------------------------- utility -----------------------------------
__global__ void fill_kernel(float* __restrict__ p, float v, int n) {
  int i = blockIdx.x * blockDim.x + threadIdx.x;
  if (i < n) p[i] = v;
}

// ------------------------------ batch norm ---------------------------------
template <int COLS>
__global__ void bn_stats_kernel(const float* __restrict__ x, int n,
                                float* __restrict__ sums) {
  __shared__ float ss[2 * COLS];
  if (threadIdx.x < 2 * COLS) ss[threadIdx.x] = 0.0f;
  __syncthreads();
  float ls[COLS], lq[COLS];
#pragma unroll
  for (int c = 0; c < COLS; ++c) { ls[c] = 0.0f; lq[c] = 0.0f; }
  for (int r = blockIdx.x * blockDim.x + threadIdx.x; r < n;
       r += gridDim.x * blockDim.x) {
#pragma unroll
    for (int c = 0; c < COLS; ++c) {
      float v = x[r * COLS + c];
      ls[c] += v;
      lq[c] += v * v;
    }
  }
#pragma unroll
  for (int c = 0; c < COLS; ++c) {
    atomicAdd(&ss[c], ls[c]);
    atomicAdd(&ss[COLS + c], lq[c]);
  }
  __syncthreads();
  if (threadIdx.x < 2 * COLS) atomicAdd(&sums[threadIdx.x], ss[threadIdx.x]);
}

__global__ void bn_finalize_kernel(const float* __restrict__ sums,
                                   const float* __restrict__ g,
                                   const float* __restrict__ b,
                                   float* __restrict__ scsh, float count,
                                   int cols) {
  int c = threadIdx.x;
  if (c >= cols) return;
  float mean = sums[c] / count;
  float var = sums[cols + c] / count - mean * mean;
  float sc = g[c] * rsqrtf(var + BN_EPS);
  scsh[c] = sc;
  scsh[cols + c] = b[c] - mean * sc;
}

__global__ void bn_apply_kernel(const float* __restrict__ x,
                                const float* __restrict__ scsh,
                                float* __restrict__ y, int total, int cols) {
  int i = blockIdx.x * blockDim.x + threadIdx.x;
  if (i >= total) return;
  int c = i % cols;
  y[i] = x[i] * scsh[c] + scsh[cols + c];
}

// --------------------------- self-loop attributes ---------------------------
__global__ void loop_accum_kernel(const float* __restrict__ ea,
                                  const int* __restrict__ dst,
                                  float* __restrict__ cnt,
                                  float* __restrict__ lsum, int E) {
  int e = blockIdx.x * blockDim.x + threadIdx.x;
  if (e >= E) return;
  int d = dst[e];
  atomicAdd(&cnt[d], 1.0f);
#pragma unroll
  for (int c = 0; c < FE; ++c) atomicAdd(&lsum[d * FE + c], ea[e * FE + c]);
}

__global__ void loop_div_kernel(float* __restrict__ loop_attr,
                                const float* __restrict__ cnt, int N) {
  int i = blockIdx.x * blockDim.x + threadIdx.x;
  if (i >= N * FE) return;
  int node = i >> 2;
  loop_attr[i] = loop_attr[i] / fmaxf(cnt[node], 1.0f);
}

// ------------------------- WMMA node GEMM (f32) ----------------------------
// Y[nrows x 64] = Hm[nrows x kdim] @ W[kdim x 64] + bias
// One wave32 computes a 16x16 f32 tile via V_WMMA_F32_16X16X4_F32.
__global__ void gemm_wmma_kernel(const float* __restrict__ Hm,
                                 const float* __restrict__ W,
                                 const float* __restrict__ bias,
                                 float* __restrict__ Y, int nrows, int kdim) {
  int wave = threadIdx.x >> 5;  // 4 waves per block
  int lane = threadIdx.x & 31;
  int numRowTiles = nrows >> 4;
  int rowTile = blockIdx.x * 4 + wave;
  if (rowTile >= numRowTiles) return;  // wave-uniform: EXEC stays all-1s
  int colTile = blockIdx.y;            // 0..3 (64 cols / 16)

  int m = lane & 15;
  int khalf = (lane >> 4) << 1;  // lanes 0-15 -> K pair {0,1}; 16-31 -> {2,3}
  int row = rowTile * 16 + m;
  int col = colTile * 16 + m;

  v8f c = {};
  for (int k0 = 0; k0 < kdim; k0 += 4) {
    int ka = k0 + khalf;
    v2f a, b;
    a.x = Hm[row * kdim + ka];
    a.y = Hm[row * kdim + ka + 1];
    b.x = W[ka * HC + col];
    b.y = W[(ka + 1) * HC + col];
    c = __builtin_amdgcn_wmma_f32_16x16x4_f32(false, a, false, b, (short)0, c,
                                              false, false);
  }

  // D layout: VGPR r -> (M = r + (lane<16?0:8), N = lane&15)
  int nOut = colTile * 16 + (lane & 15);
  int mBase = rowTile * 16 + ((lane >> 4) << 3);
  float bcol = bias[nOut];
#pragma unroll
  for (int r = 0; r < 8; ++r) {
    Y[(mBase + r) * HC + nOut] = c[r] + bcol;
  }
}

// ------------------------------ edge phase ---------------------------------
// Pass 1: per-edge logits (fused ea@We, leaky, attention dot) + segment max.
__global__ void edge_logits_kernel(const float* __restrict__ xl,
                                   const float* __restrict__ xr,
                                   const int* __restrict__ src,
                                   const int* __restrict__ dst,
                                   const float* __restrict__ ea,
                                   const float* __restrict__ loop_attr,
                                   const float* __restrict__ We,
                                   const float* __restrict__ att,
                                   float* __restrict__ logits,
                                   float* __restrict__ amax, int E, int ET) {
  __shared__ float sWe[FE * HC];
  __shared__ float sAtt[HC];
  for (int i = threadIdx.x; i < FE * HC; i += blockDim.x) sWe[i] = We[i];
  for (int i = threadIdx.x; i < HC; i += blockDim.x) sAtt[i] = att[i];
  __syncthreads();

  int e = blockIdx.x * blockDim.x + threadIdx.x;
  if (e >= ET) return;
  int s, d;
  float a0, a1, a2, a3;
  if (e < E) {
    s = src[e];
    d = dst[e];
    a0 = ea[e * FE + 0]; a1 = ea[e * FE + 1];
    a2 = ea[e * FE + 2]; a3 = ea[e * FE + 3];
  } else {
    s = e - E;
    d = s;
    const float* la = &loop_attr[(size_t)s * FE];
    a0 = la[0]; a1 = la[1]; a2 = la[2]; a3 = la[3];
  }
  const float* xls = &xl[(size_t)s * HC];
  const float* xrd = &xr[(size_t)d * HC];
#pragma unroll
  for (int h = 0; h < HEADS; ++h) {
    float acc = 0.0f;
#pragma unroll
    for (int ci = 0; ci < CPH; ++ci) {
      int j = h * CPH + ci;
      float eej = a0 * sWe[j] + a1 * sWe[HC + j] + a2 * sWe[2 * HC + j] +
                  a3 * sWe[3 * HC + j];
      acc += sAtt[j] * leaky(xls[j] + xrd[j] + eej);
    }
    logits[(size_t)e * HEADS + h] = acc;
    atomicMaxF(&amax[d * HEADS + h], acc);
  }
}

// Pass 2: w = exp(logit - amax[dst]); den[dst] += w. (logits overwritten by w)
__global__ void edge_softmax_den_kernel(float* __restrict__ wbuf,
                                        const int* __restrict__ dst,
                                        const float* __restrict__ amax,
                                        float* __restrict__ den, int E,
                                        int ET) {
  int t = blockIdx.x * blockDim.x + threadIdx.x;
  if (t >= ET * HEADS) return;
  int e = t >> 3;
  int h = t & 7;
  int d = (e < E) ? dst[e] : (e - E);
  float w = __expf(wbuf[t] - amax[d * HEADS + h]);
  wbuf[t] = w;
  atomicAdd(&den[d * HEADS + h], w);
}

// Pass 3: out[dst] += (w/den[dst]) * xl[src], one thread per (edge, head).
__global__ void edge_agg_kernel(const float* __restrict__ xl,
                                const float* __restrict__ wbuf,
                                const float* __restrict__ den,
                                const int* __restrict__ src,
                                const int* __restrict__ dst,
                                float* __restrict__ out, int E, int ET) {
  int t = blockIdx.x * blockDim.x + threadIdx.x;
  if (t >= ET * HEADS) return;
  int e = t >> 3;
  int h = t & 7;
  int s = (e < E) ? src[e] : (e - E);
  int d = (e < E) ? dst[e] : (e - E);
  float alpha = wbuf[t] / den[d * HEADS + h];
  const float* xs = &xl[(size_t)s * HC + h * CPH];
  float* od = &out[(size_t)d * HC + h * CPH];
#pragma unroll
  for (int ci = 0; ci < CPH; ++ci) atomicAdd(&od[ci], alpha * xs[ci]);
}

// Pass 4: h_out = leaky(acc + bc), in place.
__global__ void node_finish_kernel(float* __restrict__ acc,
                                   const float* __restrict__ bc, int total) {
  int t = blockIdx.x * blockDim.x + threadIdx.x;
  if (t >= total) return;
  acc[t] = leaky(acc[t] + bc[t & (HC - 1)]);
}

// ------------------------------ readout ------------------------------------
__global__ void graph_max_kernel(const float* __restrict__ h,
                                 const int* __restrict__ batch,
                                 const float* __restrict__ tscale,
                                 float* __restrict__ gmax, int total) {
  int i = blockIdx.x * blockDim.x + threadIdx.x;
  if (i >= total) return;
  int node = i >> 6;
  int c = i & (HC - 1);
  atomicMaxF(&gmax[batch[node] * HC + c], h[i] * tscale[0]);
}

__global__ void graph_den_kernel(const float* __restrict__ h,
                                 const int* __restrict__ batch,
                                 const float* __restrict__ tscale,
                                 const float* __restrict__ gmax,
                                 float* __restrict__ gden, int total) {
  int i = blockIdx.x * blockDim.x + threadIdx.x;
  if (i >= total) return;
  int node = i >> 6;
  int c = i & (HC - 1);
  int g = batch[node];
  float w = __expf(h[i] * tscale[0] - gmax[g * HC + c]);
  atomicAdd(&gden[g * HC + c], w);
}

__global__ void graph_agg_kernel(const float* __restrict__ h,
                                 const int* __restrict__ batch,
                                 const float* __restrict__ tscale,
                                 const float* __restrict__ gmax,
                                 const float* __restrict__ gden,
                                 float* __restrict__ gagg, int total) {
  int i = blockIdx.x * blockDim.x + threadIdx.x;
  if (i >= total) return;
  int node = i >> 6;
  int c = i & (HC - 1);
  int g = batch[node];
  float w = __expf(h[i] * tscale[0] - gmax[g * HC + c]);
  atomicAdd(&gagg[g * HC + c], (w / gden[g * HC + c]) * h[i]);
}

__global__ void final_linear_kernel(const float* __restrict__ gagg,
                                    const float* __restrict__ Wo,
                                    const float* __restrict__ bo,
                                    float* __restrict__ out) {
  int t = threadIdx.x;
  if (t >= NGRAPH * NCLS) return;
  int g = t >> 2;
  int k = t & 3;
  float acc = bo[k];
#pragma unroll 8
  for (int c = 0; c < HC; ++c) acc += gagg[g * HC + c] * Wo[c * NCLS + k];
  out[t] = acc;
}

// ---------------------------------------------------------------------------
extern "C" void kernel_launch(void* const* d_in, const int* in_sizes, int n_in,
                              void* d_out, int out_size, void* d_ws,
                              size_t ws_size, hipStream_t stream) {
  (void)n_in; (void)out_size; (void)ws_size;

  const float* x = (const float*)d_in[0];
  const int* edge_index = (const int*)d_in[1];
  const float* edge_attr = (const float*)d_in[2];
  const int* batch = (const int*)d_in[3];
  const float* g_n = (const float*)d_in[4];
  const float* b_n = (const float*)d_in[5];
  const float* g_e = (const float*)d_in[6];
  const float* b_e = (const float*)d_in[7];
  const float* tscale = (const float*)d_in[29];
  const float* W_out = (const float*)d_in[30];
  const float* b_out = (const float*)d_in[31];
  float* out = (float*)d_out;

  const int Fdim = 16;
  const int N = in_sizes[0] / Fdim;
  const int E = in_sizes[1] / 2;
  const int ET = E + N;
  const int* src = edge_index;
  const int* dst = edge_index + E;

  // ---- workspace layout (floats) ----
  float* ws = (float*)d_ws;
  size_t off = 0;
  float* stats = ws + off; off += 256;            // [0..31] x sums, [32..39] ea sums,
                                                  // [40..71] x scale/shift, [72..79] ea scale/shift
  float* cnt = ws + off; off += (size_t)N;
  float* loopA = ws + off; off += (size_t)N * FE;
  float* eaN = ws + off; off += (size_t)E * FE;
  float* hA = ws + off; off += (size_t)N * HC;    // layer input ping
  float* hB = ws + off; off += (size_t)N * HC;    // layer output pong
  float* xlb = ws + off; off += (size_t)N * HC;
  float* xrb = ws + off; off += (size_t)N * HC;
  float* wbuf = ws + off; off += (size_t)ET * HEADS;
  float* amax = ws + off; off += (size_t)N * HEADS;
  float* den = ws + off; off += (size_t)N * HEADS;
  float* gmax = ws + off; off += NGRAPH * HC;
  float* gden = ws + off; off += NGRAPH * HC;
  float* gagg = ws + off; off += NGRAPH * HC;

  const int B = 256;
  auto blocks = [](int n, int b) { return (n + b - 1) / b; };

  // ---- batch norm of x and edge_attr ----
  fill_kernel<<<1, 256, 0, stream>>>(stats, 0.0f, 256);
  bn_stats_kernel<16><<<512, B, 0, stream>>>(x, N, stats);
  bn_stats_kernel<4><<<512, B, 0, stream>>>(edge_attr, E, stats + 32);
  bn_finalize_kernel<<<1, 32, 0, stream>>>(stats, g_n, b_n, stats + 40,
                                           (float)N, 16);
  bn_finalize_kernel<<<1, 32, 0, stream>>>(stats + 32, g_e, b_e, stats + 72,
                                           (float)E, 4);
  bn_apply_kernel<<<blocks(N * 16, B), B, 0, stream>>>(x, stats + 40, hA,
                                                       N * 16, 16);
  bn_apply_kernel<<<blocks(E * FE, B), B, 0, stream>>>(edge_attr, stats + 72,
                                                       eaN, E * FE, FE);

  // ---- self-loop attributes: segment mean of incoming edge attrs ----
  fill_kernel<<<blocks(N, B), B, 0, stream>>>(cnt, 0.0f, N);
  fill_kernel<<<blocks(N * FE, B), B, 0, stream>>>(loopA, 0.0f, N * FE);
  loop_accum_kernel<<<blocks(E, B), B, 0, stream>>>(eaN, dst, cnt, loopA, E);
  loop_div_kernel<<<blocks(N * FE, B), B, 0, stream>>>(loopA, cnt, N);

  // ---- 3 GATv2 layers ----
  const int numRowTiles = N >> 4;  // 6250
  dim3 gemmGrid(blocks(numRowTiles, 4), 4);
  float* h_in = hA;
  float* h_out = hB;
  for (int L = 0; L < 3; ++L) {
    const float* Wl = (const float*)d_in[8 + 7 * L + 0];
    const float* bl = (const float*)d_in[8 + 7 * L + 1];
    const float* Wr = (const float*)d_in[8 + 7 * L + 2];
    const float* br = (const float*)d_in[8 + 7 * L + 3];
    const float* We = (const float*)d_in[8 + 7 * L + 4];
    const float* att = (const float*)d_in[8 + 7 * L + 5];
    const float* bc = (const float*)d_in[8 + 7 * L + 6];
    const int kdim = (L == 0) ? Fdim : HC;

    fill_kernel<<<blocks(N * HEADS, B), B, 0, stream>>>(amax, -3.0e38f,
                                                        N * HEADS);
    fill_kernel<<<blocks(N * HEADS, B), B, 0, stream>>>(den, 0.0f, N * HEADS);
    fill_kernel<<<blocks(N * HC, B), B, 0, stream>>>(h_out, 0.0f, N * HC);

    gemm_wmma_kernel<<<gemmGrid, 128, 0, stream>>>(h_in, Wl, bl, xlb, N, kdim);
    gemm_wmma_kernel<<<gemmGrid, 128, 0, stream>>>(h_in, Wr, br, xrb, N, kdim);

    edge_logits_kernel<<<blocks(ET, B), B, 0, stream>>>(
        xlb, xrb, src, dst, eaN, loopA, We, att, wbuf, amax, E, ET);
    edge_softmax_den_kernel<<<blocks(ET * HEADS, B), B, 0, stream>>>(
        wbuf, dst, amax, den, E, ET);
    edge_agg_kernel<<<blocks(ET * HEADS, B), B, 0, stream>>>(
        xlb, wbuf, den, src, dst, h_out, E, ET);
    node_finish_kernel<<<blocks(N * HC, B), B, 0, stream>>>(h_out, bc, N * HC);

    float* tmp = h_in;
    h_in = h_out;
    h_out = tmp;
  }
  float* h_fin = h_in;  // output of layer 3 after swap

  // ---- per-graph softmax readout ----
  fill_kernel<<<blocks(NGRAPH * HC, B), B, 0, stream>>>(gmax, -3.0e38f,
                                                        NGRAPH * HC);
  fill_kernel<<<blocks(NGRAPH * HC, B), B, 0, stream>>>(gden, 0.0f,
                                                        NGRAPH * HC);
  fill_kernel<<<blocks(NGRAPH * HC, B), B, 0, stream>>>(gagg, 0.0f,
                                                        NGRAPH * HC);
  graph_max_kernel<<<blocks(N * HC, B), B, 0, stream>>>(h_fin, batch, tscale,
                                                        gmax, N * HC);
  graph_den_kernel<<<blocks(N * HC, B), B, 0, stream>>>(h_fin, batch, tscale,
                                                        gmax, gden, N * HC);
  graph_agg_kernel<<<blocks(N * HC, B), B, 0, stream>>>(h_fin, batch, tscale,
                                                        gmax, gden, gagg,
                                                        N * HC);
  final_linear_kernel<<<1, 256, 0, stream>>>(gagg, W_out, b_out, out);
}